// MAGNOEncoder_46428596470308
// MI455X (gfx1250) — compile-verified
//
#include <hip/hip_runtime.h>
#include <hip/hip_bf16.h>

typedef __attribute__((ext_vector_type(2))) float v2f;
typedef __attribute__((ext_vector_type(8))) float v8f;

#define BM 128
#define BN 128
#define BK 32
#define LDA 34   // even pad: 8B-aligned v2f fragment loads, bank-conflict free
#define LDB 34   // B tile stored transposed: BsT[col][k]

// C[M,N] = A[M,K] * B[K,N] (+ bias[N]);  M % 128 == 0, N % 128 == 0, K % 32 == 0
__global__ __launch_bounds__(256) void gemm_f32_wmma(
    const float* __restrict__ A, const float* __restrict__ B,
    const float* __restrict__ bias, float* __restrict__ C,
    int M, int N, int K)
{
    __shared__ float As [BM * LDA];   // 128x32 (row-major, padded)
    __shared__ float BsT[BN * LDB];   // 32x128 stored as [col][k]

    const int tid  = threadIdx.x;
    const int wid  = tid >> 5;
    const int lane = tid & 31;
    const int bm   = blockIdx.x * BM;
    const int bn   = blockIdx.y * BN;
    const int wm   = (wid >> 1) * 32;       // wave M offset: 0,32,64,96
    const int wn   = (wid & 1) * 64;        // wave N offset: 0,64
    const int lr   = lane & 15;             // row (A) / col (B) within 16
    const int kk   = (lane >> 4) * 2;       // lanes 16..31 hold K+2, K+3

    v8f acc[2][4] = {};

    for (int k0 = 0; k0 < K; k0 += BK) {
        // Stage A tile: 128x32 floats, 16 per thread, coalesced on K
        #pragma unroll
        for (int i = 0; i < (BM * BK) / 256; ++i) {
            int idx = tid + i * 256;
            int r = idx >> 5, c = idx & 31;
            As[r * LDA + c] = A[(size_t)(bm + r) * K + k0 + c];
        }
        // Stage B tile transposed: 32x128 floats, 16 per thread, coalesced on N
        #pragma unroll
        for (int i = 0; i < (BK * BN) / 256; ++i) {
            int idx = tid + i * 256;
            int r = idx >> 7, c = idx & 127;
            BsT[c * LDB + r] = B[(size_t)(k0 + r) * N + bn + c];
        }
        // Prefetch next K-tile (each 64B chunk covered exactly once by the block)
        if (k0 + BK < K) {
            __builtin_prefetch(&A[(size_t)(bm + (tid >> 1)) * K + k0 + BK + (tid & 1) * 16], 0, 3);
            __builtin_prefetch(&B[(size_t)(k0 + BK + (tid >> 3)) * N + bn + (tid & 7) * 16], 0, 3);
        }
        __syncthreads();

        #pragma unroll
        for (int ks = 0; ks < BK; ks += 4) {
            v2f a[2], b[4];
            #pragma unroll
            for (int t = 0; t < 2; ++t)
                a[t] = *(const v2f*)(As + (wm + t * 16 + lr) * LDA + ks + kk);
            #pragma unroll
            for (int t = 0; t < 4; ++t)
                b[t] = *(const v2f*)(BsT + (wn + t * 16 + lr) * LDB + ks + kk);
            #pragma unroll
            for (int i = 0; i < 2; ++i)
                #pragma unroll
                for (int j = 0; j < 4; ++j)
                    acc[i][j] = __builtin_amdgcn_wmma_f32_16x16x4_f32(
                        false, a[i], false, b[j], (short)0, acc[i][j], false, false);
        }
        __syncthreads();
    }

    // Epilogue: C/D layout — VGPR r: lanes 0-15 -> M=r, lanes 16-31 -> M=8+r
    const int hi8 = (lane >> 4) * 8;
    #pragma unroll
    for (int i = 0; i < 2; ++i) {
        int rowbase = bm + wm + i * 16 + hi8;
        #pragma unroll
        for (int j = 0; j < 4; ++j) {
            int col = bn + wn + j * 16 + lr;
            float bv = bias ? bias[col] : 0.0f;
            #pragma unroll
            for (int r = 0; r < 8; ++r)
                C[(size_t)(rowbase + r) * N + col] = acc[i][j][r] + bv;
        }
    }
}

// scores[e,h] = dot(Qf[q_idx[e], h*32:+32], Kf[s_idx[e], h*32:+32]) / (sqrt(32)*tau)
__global__ void score_kernel(const float* __restrict__ Qf, const float* __restrict__ Kf,
                             const int* __restrict__ q_idx, const int* __restrict__ s_idx,
                             const float* __restrict__ log_tau,
                             float* __restrict__ scores, int EH)
{
    int t = blockIdx.x * blockDim.x + threadIdx.x;
    if (t >= EH) return;
    int e = t >> 3, h = t & 7;
    int q = q_idx[e], s = s_idx[e];
    const float4* qp = (const float4*)(Qf + (size_t)q * 256 + h * 32);
    const float4* kp = (const float4*)(Kf + (size_t)s * 256 + h * 32);
    float acc = 0.f;
    #pragma unroll
    for (int i = 0; i < 8; ++i) {
        float4 a = qp[i], b = kp[i];
        acc += a.x * b.x + a.y * b.y + a.z * b.z + a.w * b.w;
    }
    float scale = 0.17677669529663687f * __expf(-log_tau[0]);  // 1/sqrt(32) * 1/tau
    scores[t] = acc * scale;
}

__global__ void zero_kernel(float* __restrict__ p, int n)
{
    int t = blockIdx.x * blockDim.x + threadIdx.x;
    if (t < n) p[t] = 0.0f;
}

// segment max clamped at 0: smax starts at 0.0f bits; uint ordering == float
// ordering for non-negative floats, so only push positive scores.
__global__ void segmax_kernel(const float* __restrict__ scores, const int* __restrict__ q_idx,
                              unsigned int* __restrict__ smax, int EH)
{
    int t = blockIdx.x * blockDim.x + threadIdx.x;
    if (t >= EH) return;
    int e = t >> 3, h = t & 7;
    float sc = scores[t];
    if (sc > 0.0f)
        atomicMax(&smax[(size_t)q_idx[e] * 8 + h], __float_as_uint(sc));
}

__global__ void exsum_kernel(float* __restrict__ scores, const int* __restrict__ q_idx,
                             const float* __restrict__ smax, float* __restrict__ ssum, int EH)
{
    int t = blockIdx.x * blockDim.x + threadIdx.x;
    if (t >= EH) return;
    int e = t >> 3, h = t & 7;
    size_t idx = (size_t)q_idx[e] * 8 + h;
    float ex = __expf(scores[t] - smax[idx]);
    scores[t] = ex;                   // scores buffer now holds exp values
    atomicAdd(&ssum[idx], ex);
}

// One wave32 per query; lane l owns dims [l*8, l*8+8), head = l/4.
__global__ __launch_bounds__(32) void agg_kernel(
    const float* __restrict__ ex, const float* __restrict__ ssum,
    const float* __restrict__ Vf, const float* __restrict__ Gf,
    const int* __restrict__ q_idx, const int* __restrict__ s_idx,
    float* __restrict__ attn, int E)
{
    int q = blockIdx.x;
    int lane = threadIdx.x;

    // lower_bound(q), lower_bound(q+1) over sorted q_idx
    int lo = 0, hi = E;
    while (lo < hi) { int mid = (lo + hi) >> 1; if (q_idx[mid] < q) lo = mid + 1; else hi = mid; }
    int start = lo;
    hi = E;
    while (lo < hi) { int mid = (lo + hi) >> 1; if (q_idx[mid] < q + 1) lo = mid + 1; else hi = mid; }
    int end = lo;

    int h  = lane >> 2;
    int d0 = lane * 8;
    float inv = 1.0f / fmaxf(ssum[(size_t)q * 8 + h], 1e-8f);

    float4 g0 = *(const float4*)(Gf + (size_t)q * 256 + d0);
    float4 g1 = *(const float4*)(Gf + (size_t)q * 256 + d0 + 4);
    float4 a0 = {0.f, 0.f, 0.f, 0.f}, a1 = {0.f, 0.f, 0.f, 0.f};

    for (int e = start; e < end; ++e) {
        int s = s_idx[e];
        float alpha = ex[(size_t)e * 8 + h] * inv;
        float4 v0 = *(const float4*)(Vf + (size_t)s * 256 + d0);
        float4 v1 = *(const float4*)(Vf + (size_t)s * 256 + d0 + 4);
        a0.x += alpha * (v0.x + g0.x); a0.y += alpha * (v0.y + g0.y);
        a0.z += alpha * (v0.z + g0.z); a0.w += alpha * (v0.w + g0.w);
        a1.x += alpha * (v1.x + g1.x); a1.y += alpha * (v1.y + g1.y);
        a1.z += alpha * (v1.z + g1.z); a1.w += alpha * (v1.w + g1.w);
    }
    *(float4*)(attn + (size_t)q * 256 + d0)     = a0;
    *(float4*)(attn + (size_t)q * 256 + d0 + 4) = a1;
}

extern "C" void kernel_launch(void* const* d_in, const int* in_sizes, int n_in,
                              void* d_out, int out_size, void* d_ws, size_t ws_size,
                              hipStream_t stream)
{
    (void)n_in; (void)out_size; (void)ws_size;
    const float* query   = (const float*)d_in[0];
    const float* support = (const float*)d_in[1];
    const float* geo     = (const float*)d_in[2];
    const int*   q_idx   = (const int*)d_in[3];
    const int*   s_idx   = (const int*)d_in[4];
    const float* Wq      = (const float*)d_in[6];
    const float* Wk      = (const float*)d_in[7];
    const float* Wv      = (const float*)d_in[8];
    const float* Wg      = (const float*)d_in[9];
    const float* Wo      = (const float*)d_in[10];
    const float* bo      = (const float*)d_in[11];
    const float* log_tau = (const float*)d_in[12];

    const int D = 256;
    const int Q = in_sizes[0] / D;       // 8192
    const int N = in_sizes[1] / D;       // 65536
    const int E = in_sizes[3];           // 262144
    const int G = in_sizes[2] / Q;       // 64
    const int EH = E * 8;

    // workspace layout (floats)
    float* ws = (float*)d_ws;
    float* Qf   = ws;                               // Q*D
    float* Kf   = Qf   + (size_t)Q * D;             // N*D
    float* Vf   = Kf   + (size_t)N * D;             // N*D
    float* Gf   = Vf   + (size_t)N * D;             // Q*D
    float* sc   = Gf   + (size_t)Q * D;             // E*8 (scores -> exp)
    float* smax = sc   + (size_t)EH;                // Q*8
    float* ssum = smax + (size_t)Q * 8;             // Q*8
    float* attn = ssum + (size_t)Q * 8;             // Q*D

    dim3 blk(256);
    // projections (WMMA GEMMs)
    gemm_f32_wmma<<<dim3(Q / BM, D / BN), blk, 0, stream>>>(query,   Wq, nullptr, Qf, Q, D, D);
    gemm_f32_wmma<<<dim3(N / BM, D / BN), blk, 0, stream>>>(support, Wk, nullptr, Kf, N, D, D);
    gemm_f32_wmma<<<dim3(N / BM, D / BN), blk, 0, stream>>>(support, Wv, nullptr, Vf, N, D, D);
    gemm_f32_wmma<<<dim3(Q / BM, D / BN), blk, 0, stream>>>(geo,     Wg, nullptr, Gf, Q, D, G);

    // edge scores
    score_kernel<<<(EH + 255) / 256, blk, 0, stream>>>(Qf, Kf, q_idx, s_idx, log_tau, sc, EH);

    // softmax stats (smax and ssum are contiguous -> one zero pass)
    zero_kernel<<<(Q * 16 + 255) / 256, blk, 0, stream>>>(smax, Q * 16);
    segmax_kernel<<<(EH + 255) / 256, blk, 0, stream>>>(sc, q_idx, (unsigned int*)smax, EH);
    exsum_kernel<<<(EH + 255) / 256, blk, 0, stream>>>(sc, q_idx, smax, ssum, EH);

    // per-query weighted aggregation (one wave32 per query)
    agg_kernel<<<Q, 32, 0, stream>>>(sc, ssum, Vf, Gf, q_idx, s_idx, attn, E);

    // output projection with fused bias
    gemm_f32_wmma<<<dim3(Q / BM, D / BN), blk, 0, stream>>>(attn, Wo, bo, (float*)d_out, Q, D, D);
}